// DirectionFeatureExtractor_45552423141443
// MI455X (gfx1250) — compile-verified
//
#include <hip/hip_runtime.h>

#define NB 9
#define PI_F 3.14159265358979323846f

typedef __attribute__((ext_vector_type(2))) float v2f;
typedef __attribute__((ext_vector_type(8))) float v8f;

__device__ __forceinline__ int reflect512(int i) {
    if (i < 0) i = -i;
    if (i > 511) i = 1022 - i;
    return i;
}

__device__ __forceinline__ float sinc_f(float x) {
    if (x == 0.0f) return 1.0f;
    float px = PI_F * x;
    return sinf(px) / px;
}

// ---- K1: grayscale (B,3,512,512) -> (B,512,512) ----
__global__ void k_gray(const float* __restrict__ x, float* __restrict__ gray) {
    int g = blockIdx.x * blockDim.x + threadIdx.x;       // 2,097,152 threads
    int b = g >> 18;                                     // /262144
    int p = g & 262143;
    const float* xb = x + (size_t)b * 786432;
    gray[g] = 0.299f * xb[p] + 0.587f * xb[262144 + p] + 0.114f * xb[524288 + p];
}

// ---- K2: fused gradients + binning + 33x33 radial-tent cell aggregation ----
// one block (256 thr) per (b, oy, ox) cell; output hog (B,9,32,32)
__global__ void k_hog(const float* __restrict__ gray, float* __restrict__ hog) {
    __shared__ float sacc[NB][256];
    int blk = blockIdx.x;
    int b   = blk >> 10;
    int rem = blk & 1023;
    int oy  = rem >> 5, ox = rem & 31;
    const float* gb = gray + (size_t)b * 262144;

    float acc[NB];
#pragma unroll
    for (int k = 0; k < NB; k++) acc[k] = 0.0f;

    const float inv20 = 1.0f / 20.0f;
    const float invd  = 1.0f / (16.0f * 1.41421356237f);

    for (int j = threadIdx.x; j < 1089; j += 256) {
        int dy = j / 33, dx = j - dy * 33;
        float fy = (float)(dy - 16), fx = (float)(dx - 16);
        float filt = 1.0f - sqrtf(fy * fy + fx * fx) * invd;
        int y  = reflect512(oy * 16 + dy - 16);
        int xx = reflect512(ox * 16 + dx - 16);
        // zero-padded central differences on the original image
        float gl = (xx >= 1)   ? gb[y * 512 + xx - 1] : 0.0f;
        float gr = (xx <= 510) ? gb[y * 512 + xx + 1] : 0.0f;
        float gu = (y >= 1)    ? gb[(y - 1) * 512 + xx] : 0.0f;
        float gd = (y <= 510)  ? gb[(y + 1) * 512 + xx] : 0.0f;
        float gx = gr - gl, gy = gd - gu;
        float inten = sqrtf(gx * gx + gy * gy);
        float ori = fmodf(atan2f(gy, gx) + PI_F, PI_F) * (180.0f / PI_F);
        float wi = inten * filt;
#pragma unroll
        for (int k = 0; k < NB; k++) {
            float c = 10.0f + 20.0f * (float)k;
            float w = fmaxf(1.0f - fabsf(ori - c) * inv20, 0.0f);
            acc[k] += w * wi;
        }
    }
#pragma unroll
    for (int k = 0; k < NB; k++) sacc[k][threadIdx.x] = acc[k];
    __syncthreads();
    for (int s = 128; s > 0; s >>= 1) {
        if (threadIdx.x < (unsigned)s)
#pragma unroll
            for (int k = 0; k < NB; k++)
                sacc[k][threadIdx.x] += sacc[k][threadIdx.x + s];
        __syncthreads();
    }
    if (threadIdx.x < NB)
        hog[((size_t)b * NB + threadIdx.x) * 1024 + oy * 32 + ox] = sacc[threadIdx.x][0];
}

// ---- K3: per-batch max of per-pixel L2 over bins ----
__global__ void k_maxn(const float* __restrict__ hog, float* __restrict__ maxn) {
    __shared__ float sm[256];
    int b = blockIdx.x;
    float mx = 0.0f;
    for (int p = threadIdx.x; p < 1024; p += 256) {
        float ss = 0.0f;
        for (int k = 0; k < NB; k++) {
            float v = hog[((size_t)b * NB + k) * 1024 + p];
            ss += v * v;
        }
        mx = fmaxf(mx, sqrtf(ss));
    }
    sm[threadIdx.x] = mx;
    __syncthreads();
    for (int s = 128; s > 0; s >>= 1) {
        if (threadIdx.x < (unsigned)s)
            sm[threadIdx.x] = fmaxf(sm[threadIdx.x], sm[threadIdx.x + s]);
        __syncthreads();
    }
    if (threadIdx.x == 0) maxn[b] = sm[0];
}

// ---- K4: dominant direction (top eigenvector of 2x2 Gram) + floor-quantize ----
__global__ void k_dom(const float* __restrict__ hog, const float* __restrict__ maxn,
                      float* __restrict__ q) {
    int g = blockIdx.x * blockDim.x + threadIdx.x;  // 8192
    int b = g >> 10, p = g & 1023;
    float inv = 1.0f / maxn[b];
    float A = 0.0f, C = 0.0f, Bb = 0.0f;
#pragma unroll
    for (int k = 0; k < NB; k++) {
        float th = (10.0f + 20.0f * (float)k) * (PI_F / 180.0f);
        float s = sinf(th), c = cosf(th);
        float v = hog[((size_t)b * NB + k) * 1024 + p] * inv;
        float h2 = v * v;
        A += s * s * h2;
        C += c * c * h2;
        Bb -= s * c * h2;
    }
    float half = (A - C) * 0.5f;
    float lam = (A + C) * 0.5f + sqrtf(half * half + Bb * Bb);
    float n1 = Bb * Bb + (lam - A) * (lam - A);
    float n2 = (lam - C) * (lam - C) + Bb * Bb;
    float vx, vy;
    if (n1 >= n2) { vx = Bb;      vy = lam - A; }
    else          { vx = lam - C; vy = Bb;      }
    float nrm = sqrtf(vx * vx + vy * vy) + 1e-9f;
    vx /= nrm; vy /= nrm;
    q[((size_t)b * 2 + 0) * 1024 + p] = floorf((vx + 1.0f) * 0.5f * 255.0f);
    q[((size_t)b * 2 + 1) * 1024 + p] = floorf((vy + 1.0f) * 0.5f * 255.0f);
}

// ---- K5: Lanczos3 resize matrix R (512x32), row-normalized like jax.image.resize ----
__global__ void k_resizemat(float* __restrict__ R) {
    int u = blockIdx.x * blockDim.x + threadIdx.x;
    if (u >= 512) return;
    float pos = ((float)u + 0.5f) * (1.0f / 16.0f) - 0.5f;
    float w[32];
    float sum = 0.0f;
    for (int s = 0; s < 32; s++) {
        float xv = pos - (float)s;
        float v = 0.0f;
        if (fabsf(xv) < 3.0f) v = sinc_f(xv) * sinc_f(xv * (1.0f / 3.0f));
        w[s] = v; sum += v;
    }
    float invs = 1.0f / sum;
    for (int s = 0; s < 32; s++) R[u * 32 + s] = w[s] * invs;
}

// ---- K6a: tmp(512x32) = R(512x32) @ q(32x32) per image, via V_WMMA_F32_16X16X4_F32 ----
// one wave per 16x16 tile of tmp; 16 images * (32 Mtiles) * (2 Ntiles) = 1024 waves
__global__ void k_mm1(const float* __restrict__ R, const float* __restrict__ q,
                      float* __restrict__ tmp) {
    int blk = blockIdx.x;
    int bc  = blk >> 6;
    int rem = blk & 63;
    int m0 = (rem >> 1) * 16;
    int n0 = (rem & 1) * 16;
    int lane = threadIdx.x;
    int idx  = lane & 15;
    int koff = (lane >> 4) * 2;
    const float* qi = q + (size_t)bc * 1024;
    v8f acc = {};
#pragma unroll
    for (int k0 = 0; k0 < 32; k0 += 4) {
        int ka = k0 + koff;
        v2f a, bm;
        a.x  = R[(m0 + idx) * 32 + ka];
        a.y  = R[(m0 + idx) * 32 + ka + 1];
        bm.x = qi[ka * 32 + n0 + idx];
        bm.y = qi[(ka + 1) * 32 + n0 + idx];
        acc = __builtin_amdgcn_wmma_f32_16x16x4_f32(false, a, false, bm,
                                                    (short)0, acc, false, false);
    }
    int radd = (lane >> 4) * 8;
    float* ti = tmp + (size_t)bc * 16384;
#pragma unroll
    for (int r = 0; r < 8; r++)
        ti[(m0 + r + radd) * 32 + n0 + idx] = acc[r];
}

// ---- K6b: out(512x512) = tmp @ R^T, fused round/clip/[-1,1] rescale ----
// 16 images * 32 * 32 tiles = 16384 waves
__global__ void k_mm2(const float* __restrict__ tmp, const float* __restrict__ R,
                      float* __restrict__ out) {
    int blk = blockIdx.x;
    int bc  = blk >> 10;
    int rem = blk & 1023;
    int m0 = (rem >> 5) * 16;
    int n0 = (rem & 31) * 16;
    int lane = threadIdx.x;
    int idx  = lane & 15;
    int koff = (lane >> 4) * 2;
    const float* ti = tmp + (size_t)bc * 16384;
    v8f acc = {};
#pragma unroll
    for (int k0 = 0; k0 < 32; k0 += 4) {
        int ka = k0 + koff;
        v2f a, bm;
        a.x  = ti[(m0 + idx) * 32 + ka];
        a.y  = ti[(m0 + idx) * 32 + ka + 1];
        bm.x = R[(n0 + idx) * 32 + ka];      // B[k][n] = R[n][k]  (R^T)
        bm.y = R[(n0 + idx) * 32 + ka + 1];
        acc = __builtin_amdgcn_wmma_f32_16x16x4_f32(false, a, false, bm,
                                                    (short)0, acc, false, false);
    }
    int radd = (lane >> 4) * 8;
    float* ob = out + (size_t)bc * 262144;
#pragma unroll
    for (int r = 0; r < 8; r++) {
        float v = rintf(acc[r]);                     // round-half-even like jnp.round
        v = fminf(fmaxf(v, 0.0f), 255.0f);
        v = (v * (1.0f / 255.0f) - 0.5f) * 2.0f;
        ob[(m0 + r + radd) * 512 + n0 + idx] = v;
    }
}

// ---- K7: in-place per-pixel L2 normalize over the 2 channels ----
__global__ void k_norm(float* __restrict__ out) {
    int g = blockIdx.x * blockDim.x + threadIdx.x;   // 2,097,152
    int b = g >> 18;
    int p = g & 262143;
    size_t i0 = ((size_t)b * 2) * 262144 + p;
    size_t i1 = i0 + 262144;
    float a = out[i0], c = out[i1];
    float nrm = sqrtf(a * a + c * c);
    out[i0] = a / nrm;
    out[i1] = c / nrm;
}

extern "C" void kernel_launch(void* const* d_in, const int* in_sizes, int n_in,
                              void* d_out, int out_size, void* d_ws, size_t ws_size,
                              hipStream_t stream) {
    (void)in_sizes; (void)n_in; (void)out_size; (void)ws_size;
    const float* x = (const float*)d_in[0];

    float* ws   = (float*)d_ws;
    float* gray = ws;                 // 8*512*512      = 2,097,152
    float* hog  = gray + 2097152;     // 8*9*32*32      =    73,728
    float* maxn = hog  + 73728;       //                =         8
    float* qbuf = maxn + 8;           // 8*2*32*32      =    16,384
    float* Rm   = qbuf + 16384;       // 512*32         =    16,384
    float* tmp  = Rm   + 16384;       // 16*512*32      =   262,144   (~9.4 MB total)

    k_gray     <<<8192,  256, 0, stream>>>(x, gray);
    k_hog      <<<8192,  256, 0, stream>>>(gray, hog);
    k_maxn     <<<8,     256, 0, stream>>>(hog, maxn);
    k_dom      <<<32,    256, 0, stream>>>(hog, maxn, qbuf);
    k_resizemat<<<2,     256, 0, stream>>>(Rm);
    k_mm1      <<<1024,   32, 0, stream>>>(Rm, qbuf, tmp);
    k_mm2      <<<16384,  32, 0, stream>>>(tmp, Rm, (float*)d_out);
    k_norm     <<<8192,  256, 0, stream>>>((float*)d_out);
}